// HeteroGNN_75943611728726
// MI455X (gfx1250) — compile-verified
//
#include <hip/hip_runtime.h>

typedef __attribute__((ext_vector_type(2))) float v2f;
typedef __attribute__((ext_vector_type(8))) float v8f;

#define NNODE 100000
#define NEDGE 500000
#define FIN   256
#define HDIM  128

// ---------------- zero fill (float4 grid-stride) ----------------
__global__ void k_zero4(float4* __restrict__ p, long n4) {
  long i = (long)blockIdx.x * blockDim.x + threadIdx.x;
  long stride = (long)gridDim.x * blockDim.x;
  float4 z; z.x = 0.f; z.y = 0.f; z.z = 0.f; z.w = 0.f;
  for (; i < n4; i += stride) p[i] = z;
}

// ---------------- leaky relu in place ----------------
__global__ void k_leaky(float* __restrict__ p, long n) {
  long i = (long)blockIdx.x * blockDim.x + threadIdx.x;
  long stride = (long)gridDim.x * blockDim.x;
  for (; i < n; i += stride) {
    float v = p[i];
    p[i] = v > 0.f ? v : 0.01f * v;
  }
}

// ---------------- edge scatter: agg[dst] += x[src] * ew ----------------
// one wave (32 lanes) per edge; each lane handles 4 of the 128 features.
// HW fp32 atomic at device scope (agg lives in global; fully L2-resident).
__device__ __forceinline__ void atomicAddF(float* p, float v) {
  asm volatile("global_atomic_add_f32 %0, %1, off scope:SCOPE_DEV"
               :: "v"(p), "v"(v) : "memory");
}

__global__ void k_scatter(const float* __restrict__ xsrc,
                          const int* __restrict__ src,
                          const int* __restrict__ dst,
                          const float* __restrict__ ew,
                          float* __restrict__ agg, int E) {
  int gw   = (int)((blockIdx.x * blockDim.x + threadIdx.x) >> 5);
  int lane = threadIdx.x & 31;
  if (gw >= E) return;
  int   s = src[gw];
  int   d = dst[gw];
  float w = ew[gw];
  const float4 v = ((const float4*)(xsrc + (long)s * HDIM))[lane];
  float* a = agg + (long)d * HDIM + lane * 4;
  atomicAddF(a + 0, v.x * w);
  atomicAddF(a + 1, v.y * w);
  atomicAddF(a + 2, v.z * w);
  atomicAddF(a + 3, v.w * w);
}

// ---------------- fp32 WMMA GEMM, N fixed at 128 ----------------
// out[m,n] = act( (Cin? Cin[m,n]:0) + A1@W1 (+ A2@W2) + bias[n] )
// Block: 256 threads = 8 waves; each wave one 16x16 tile of a 16x128 strip.
// A tile staged in LDS (rows padded to K+4 floats: conflict-free ds_load_b64,
// keeps 16B alignment for the b128 fill stores).
__global__ void k_gemm(const float* __restrict__ A1, const float* __restrict__ W1, int K1,
                       const float* __restrict__ A2, const float* __restrict__ W2, int K2,
                       const float* __restrict__ bias, const float* __restrict__ Cin,
                       float* __restrict__ Cout, int act) {
  __shared__ float Alds[16 * (FIN + 4)];   // worst case K=256 -> 16.6 KB

  const int t    = threadIdx.x;
  const int lane = t & 31;
  const int wave = t >> 5;
  const int m0   = blockIdx.x * 16;
  const int n     = wave * 16 + (lane & 15);
  const int mbase = m0 + 8 * (lane >> 4);   // C/D: VGPR r holds M = mbase + r
  const int khi   = (lane >> 4) * 2;        // lanes 16-31 hold K=k+2,k+3

  v8f c = {};
  if (Cin) {
#pragma unroll
    for (int r = 0; r < 8; ++r) c[r] = Cin[(long)(mbase + r) * HDIM + n];
  }

  const float* As[2] = {A1, A2};
  const float* Ws[2] = {W1, W2};
  const int    Ks[2] = {K1, K2};

  for (int p = 0; p < 2; ++p) {
    const float* A = As[p];
    if (!A) break;                       // uniform across block
    const float* W = Ws[p];
    const int K = Ks[p];                 // 128 or 256 (power of two)
    const int rowStride = K + 4;
    const int kshift = (K == 256) ? 6 : 5;   // log2(K/4)
    const int kmask  = (K >> 2) - 1;

    __syncthreads();                     // protect LDS before overwrite
    // cooperative staging: 16 rows x K floats, coalesced float4 loads
    const int nf4 = (16 * K) >> 2;
    for (int idx = t; idx < nf4; idx += 256) {
      int row = idx >> kshift;
      int c4  = idx & kmask;
      const float4 vv = *(const float4*)(A + (long)(m0 + row) * K + (c4 << 2));
      *(float4*)(&Alds[row * rowStride + (c4 << 2)]) = vv;
    }
    __syncthreads();

    const int mloc = (lane & 15) * rowStride;
    for (int k = 0; k < K; k += 4) {
      const int kk = k + khi;
      v2f a = *(const v2f*)(&Alds[mloc + kk]);      // ds_load_b64
      v2f b;
      b.x = W[(long)kk * HDIM + n];                 // B[kk][n]
      b.y = W[(long)(kk + 1) * HDIM + n];           // B[kk+1][n]
      c = __builtin_amdgcn_wmma_f32_16x16x4_f32(false, a, false, b,
                                                (short)0, c, false, false);
    }
  }

  const float bv = bias ? bias[n] : 0.f;
#pragma unroll
  for (int r = 0; r < 8; ++r) {
    float v = c[r] + bv;
    if (act == 1)      v = v > 0.f ? v : 0.f;          // relu
    else if (act == 2) v = v > 0.f ? v : 0.01f * v;    // leaky relu
    Cout[(long)(mbase + r) * HDIM + n] = v;
  }
}

// ---------------- host-side orchestration ----------------
extern "C" void kernel_launch(void* const* d_in, const int* in_sizes, int n_in,
                              void* d_out, int out_size, void* d_ws, size_t ws_size,
                              hipStream_t stream) {
  // node features (insertion order of setup_inputs)
  const float* xin[3] = {(const float*)d_in[0], (const float*)d_in[1], (const float*)d_in[2]};
  // edges in data order: has, sug, ind, isb  (node type: P=0, I=1, T=2)
  const int*   esrc[4]; const int* edst[4]; const float* eew[4];
  for (int e = 0; e < 4; ++e) {
    esrc[e] = (const int*)d_in[3 + 3 * e];
    edst[e] = (const int*)d_in[4 + 3 * e];
    eew[e]  = (const float*)d_in[5 + 3 * e];
  }
  const int est[4] = {0, 0, 1, 2};   // has: P->, sug: P->, ind: I->, isb: T->
  const int edt[4] = {1, 2, 0, 0};   // has: ->I, sug: ->T, ind: ->P, isb: ->P

  // ---- parameter leaves: support both insertion-order and jax sorted-key order ----
  const float* linW[3];  const float* linB[3];
  const float* Wrel[2][4]; const float* brel[2][4]; const float* Wroot[2][4];
  const float* outW = nullptr; const float* outB = nullptr;
  int pi = 15;
  if (in_sizes[pi] == FIN * HDIM) {
    // insertion order: lin{P,I,T:(W,b)}, convs{l0,l1: has,sug,ind,isb:(Wrel,brel,Wroot)}, out(W,b)
    for (int t = 0; t < 3; ++t) { linW[t] = (const float*)d_in[pi++]; linB[t] = (const float*)d_in[pi++]; }
    for (int l = 0; l < 2; ++l)
      for (int e = 0; e < 4; ++e) {
        Wrel[l][e]  = (const float*)d_in[pi++];
        brel[l][e]  = (const float*)d_in[pi++];
        Wroot[l][e] = (const float*)d_in[pi++];
      }
    outW = (const float*)d_in[pi++]; outB = (const float*)d_in[pi++];
  } else {
    // jax pytree sorted keys: convs first (has,ind,isb,sug : Wrel,Wroot,brel),
    // then lin (impression,protocol,treatment : W,b), then out (W,b)
    const int se2e[4] = {0, 2, 3, 1};  // sorted edge name -> data edge index
    for (int l = 0; l < 2; ++l)
      for (int s = 0; s < 4; ++s) {
        int e = se2e[s];
        Wrel[l][e]  = (const float*)d_in[pi++];
        Wroot[l][e] = (const float*)d_in[pi++];
        brel[l][e]  = (const float*)d_in[pi++];
      }
    const int sn2n[3] = {1, 0, 2};     // impression, protocol, treatment -> node index
    for (int s = 0; s < 3; ++s) {
      int nt = sn2n[s];
      linW[nt] = (const float*)d_in[pi++];
      linB[nt] = (const float*)d_in[pi++];
    }
    outW = (const float*)d_in[pi++]; outB = (const float*)d_in[pi++];
  }

  // ---- workspace: x[3] | new[3] | agg, each NNODE*HDIM fp32 (~358 MB total) ----
  float* ws = (float*)d_ws;
  const long NB = (long)NNODE * HDIM;
  float* xb[3] = {ws,          ws + NB,     ws + 2 * NB};
  float* nb[3] = {ws + 3 * NB, ws + 4 * NB, ws + 5 * NB};
  float* agg   =  ws + 6 * NB;

  const int gemmGrid = NNODE / 16;   // 6250, exact
  // input projection + relu
  for (int t = 0; t < 3; ++t)
    k_gemm<<<gemmGrid, 256, 0, stream>>>(xin[t], linW[t], FIN,
                                         nullptr, nullptr, 0,
                                         linB[t], nullptr, xb[t], /*relu*/1);

  for (int l = 0; l < 2; ++l) {
    k_zero4<<<2048, 256, 0, stream>>>((float4*)nb[0], (3 * NB) / 4);
    for (int e = 0; e < 4; ++e) {
      k_zero4<<<2048, 256, 0, stream>>>((float4*)agg, NB / 4);
      k_scatter<<<NEDGE / 8, 256, 0, stream>>>(xb[est[e]], esrc[e], edst[e], eew[e],
                                               agg, NEDGE);
      // n[dt] += agg @ Wrel + x[dt] @ Wroot + brel
      k_gemm<<<gemmGrid, 256, 0, stream>>>(agg, Wrel[l][e], HDIM,
                                           xb[edt[e]], Wroot[l][e], HDIM,
                                           brel[l][e], nb[edt[e]], nb[edt[e]], /*none*/0);
    }
    k_leaky<<<4096, 256, 0, stream>>>(nb[0], 3 * NB);
    for (int t = 0; t < 3; ++t) { float* tmp = xb[t]; xb[t] = nb[t]; nb[t] = tmp; }
  }

  // output head: protocol features @ outW + outB
  k_gemm<<<gemmGrid, 256, 0, stream>>>(xb[0], outW, HDIM,
                                       nullptr, nullptr, 0,
                                       outB, nullptr, (float*)d_out, /*none*/0);
}